// PointSample_69028714381455
// MI455X (gfx1250) — compile-verified
//
#include <hip/hip_runtime.h>
#include <hip/hip_bf16.h>

// PointSample (bilinear grid_sample, align_corners=False, border clamp)
// features: [B, H, W, C] fp32 channels-last; grid: [B, P, 2] in [0,1]
// out: [B, P, C] fp32.
//
// MI455X strategy: memory-bound gather. One wave32 per point; 32 lanes x
// float4 covers C=128 exactly -> each corner is one coalesced 512B b128
// burst. Point-uniform state is scalarized via readfirstlane so grid reads
// and address math ride the scalar path. Output is streamed with
// non-temporal b128 stores to preserve L2 for the gathered features.

#define PS_B 8
#define PS_H 256
#define PS_W 256
#define PS_C 128
#define PS_P 8192
#define PS_NPOINTS (PS_B * PS_P)

typedef float __attribute__((ext_vector_type(4))) v4f;

__global__ __launch_bounds__(256) void PointSample_kernel(
    const float* __restrict__ features,
    const float* __restrict__ grid,
    float* __restrict__ out)
{
    const int lane = threadIdx.x & 31;        // channel chunk: lane*4 .. lane*4+3
    const int wave = threadIdx.x >> 5;        // 8 waves (points) per block
    int q = (int)blockIdx.x * 8 + wave;       // global point index
    if (q >= PS_NPOINTS) return;

    // q is uniform across the wave -> force scalar so the grid fetch and
    // all address arithmetic go down the SALU/SMEM path (wave32).
    q = __builtin_amdgcn_readfirstlane(q);

    const int b = q >> 13;                    // q / PS_P
    // point coords (wave-uniform 8B read)
    const float gx = grid[(size_t)q * 2 + 0];
    const float gy = grid[(size_t)q * 2 + 1];

    // Replicate reference math sequence exactly.
    const float xg = gx * 2.0f - 1.0f;
    const float yg = gy * 2.0f - 1.0f;
    const float ix = ((xg + 1.0f) * (float)PS_W - 1.0f) * 0.5f;
    const float iy = ((yg + 1.0f) * (float)PS_H - 1.0f) * 0.5f;

    const float x0f = floorf(ix);
    const float y0f = floorf(iy);
    const float wx = ix - x0f;
    const float wy = iy - y0f;

    const int x0i = (int)x0f;
    const int y0i = (int)y0f;
    const int x0 = min(max(x0i,     0), PS_W - 1);
    const int x1 = min(max(x0i + 1, 0), PS_W - 1);
    const int y0 = min(max(y0i,     0), PS_H - 1);
    const int y1 = min(max(y0i + 1, 0), PS_H - 1);

    const float w00 = (1.0f - wy) * (1.0f - wx);
    const float w01 = (1.0f - wy) * wx;
    const float w10 = wy * (1.0f - wx);
    const float w11 = wy * wx;

    // Corner row bases (wave-uniform); lane supplies the 16B channel offset.
    const size_t bBase = (size_t)b * (PS_H * PS_W * PS_C);
    const float* p00 = features + bBase + ((size_t)y0 * PS_W + x0) * PS_C;
    const float* p01 = features + bBase + ((size_t)y0 * PS_W + x1) * PS_C;
    const float* p10 = features + bBase + ((size_t)y1 * PS_W + x0) * PS_C;
    const float* p11 = features + bBase + ((size_t)y1 * PS_W + x1) * PS_C;

    // Four coalesced 512B gathers (global_load_b128 per lane).
    const v4f f00 = *(const v4f*)(p00 + lane * 4);
    const v4f f01 = *(const v4f*)(p01 + lane * 4);
    const v4f f10 = *(const v4f*)(p10 + lane * 4);
    const v4f f11 = *(const v4f*)(p11 + lane * 4);

    v4f acc = f00 * w00;
    acc += f01 * w01;
    acc += f10 * w10;
    acc += f11 * w11;

    // Streaming write-once output: non-temporal b128 store.
    v4f* dst = (v4f*)(out + (size_t)q * PS_C) + lane;
    __builtin_nontemporal_store(acc, dst);
}

extern "C" void kernel_launch(void* const* d_in, const int* in_sizes, int n_in,
                              void* d_out, int out_size, void* d_ws, size_t ws_size,
                              hipStream_t stream)
{
    (void)in_sizes; (void)n_in; (void)d_ws; (void)ws_size; (void)out_size;
    const float* features = (const float*)d_in[0];
    const float* grid     = (const float*)d_in[1];
    float* out            = (float*)d_out;

    // 8 points (waves) per 256-thread block.
    const int blocks = (PS_NPOINTS + 7) / 8;
    hipLaunchKernelGGL(PointSample_kernel, dim3(blocks), dim3(256), 0, stream,
                       features, grid, out);
}